// ROIPooling2D_2637109920550
// MI455X (gfx1250) — compile-verified
//
#include <hip/hip_runtime.h>
#include <stdint.h>

// ROI crop-and-resize (bilinear) + 2x2 max pool, NHWC fp32.
// feature: (1, 50, 50, 512) f32 ; rois: (N,4) f32 ; img_size: (2) i32
// out: (N, 7, 7, 512) f32
//
// One 256-thread block per (roi, pooled_row). 28 sample points (2 crop rows
// x 14 cols); each point needs 4 corner pixel vectors of 512 floats (8 KB).
// Corners are staged into LDS via CDNA5 async global->LDS loads (ASYNCcnt
// path) with a 4-deep pipeline (prefetch distance 3) to hide L2 latency.
// Each thread stages & consumes exactly its own 2 channels -> no workgroup
// barrier needed; async loads complete in order, so s_wait_asynccnt alone
// provides the data-ready guarantee.

constexpr int FH = 50, FW = 50, FC = 512;   // feature dims
constexpr int CH = 14, CW = 14;             // crop grid
constexpr int PH = 7,  PW = 7;              // pooled grid
constexpr int THREADS = 256;
constexpr int CPT = FC / THREADS;           // 2 channels per thread
constexpr int NP  = 2 * CW;                 // 28 sample points per block
constexpr int NBUF = 4;                     // pipeline depth (LDS buffers)
constexpr int PF   = NBUF - 1;              // prefetch distance

__device__ __forceinline__ void wait_async(int n) {
    // n is a compile-time constant after full unroll; branches fold away.
    if (n >= 12)     asm volatile("s_wait_asynccnt 0xc" ::: "memory");
    else if (n == 8) asm volatile("s_wait_asynccnt 0x8" ::: "memory");
    else if (n == 4) asm volatile("s_wait_asynccnt 0x4" ::: "memory");
    else             asm volatile("s_wait_asynccnt 0x0" ::: "memory");
}

__global__ __launch_bounds__(THREADS)
void roi_pool_async_kernel(const float* __restrict__ feature,
                           const float* __restrict__ rois,
                           const int*   __restrict__ img_size,
                           float*       __restrict__ out)
{
    __shared__ float sh[NBUF][4][FC];       // [buf][corner][channel] = 32 KB

    const int blk = blockIdx.x;
    const int roi = blk / PH;
    const int pyr = blk % PH;               // pooled row
    const int tid = (int)threadIdx.x;
    const int c0  = tid * CPT;

    const float ih = (float)img_size[0];
    const float iw = (float)img_size[1];

    const float4 rv = reinterpret_cast<const float4*>(rois)[roi];
    const float x1 = rv.x / iw;
    const float y1 = rv.y / ih;
    const float x2 = rv.z / iw;
    const float y2 = rv.w / ih;

    const float oy = y1 * (float)(FH - 1);
    const float ox = x1 * (float)(FW - 1);
    const float sy = (y2 - y1) * (float)(FH - 1) / (float)(CH - 1);
    const float sx = (x2 - x1) * (float)(FW - 1) / (float)(CW - 1);

    const uint32_t lds_base = (uint32_t)(uintptr_t)&sh[0][0][0];
    const unsigned long long gbase = (unsigned long long)(uintptr_t)feature;

    // Issue the 4 corner async copies for sample point p into LDS buffer `buf`.
    auto issue = [&](int p, int buf) {
        const int gy = 2 * pyr + (p / CW);
        const int gx = p % CW;
        const float in_y = oy + sy * (float)gy;
        const float in_x = ox + sx * (float)gx;
        const int top  = (int)floorf(in_y);
        const int left = (int)floorf(in_x);
        const int t = min(max(top,      0), FH - 1);
        const int b = min(max(top + 1,  0), FH - 1);
        const int l = min(max(left,     0), FW - 1);
        const int r = min(max(left + 1, 0), FW - 1);

        const uint32_t o_tl = (uint32_t)(((t * FW + l) * FC + c0) * 4);
        const uint32_t o_tr = (uint32_t)(((t * FW + r) * FC + c0) * 4);
        const uint32_t o_bl = (uint32_t)(((b * FW + l) * FC + c0) * 4);
        const uint32_t o_br = (uint32_t)(((b * FW + r) * FC + c0) * 4);

        const uint32_t dbase = lds_base + (uint32_t)(buf * 4 * FC + c0) * 4u;
        const uint32_t d_tl = dbase + 0u * FC * 4u;
        const uint32_t d_tr = dbase + 1u * FC * 4u;
        const uint32_t d_bl = dbase + 2u * FC * 4u;
        const uint32_t d_br = dbase + 3u * FC * 4u;

        asm volatile("global_load_async_to_lds_b64 %0, %1, %2"
                     :: "v"(d_tl), "v"(o_tl), "s"(gbase) : "memory");
        asm volatile("global_load_async_to_lds_b64 %0, %1, %2"
                     :: "v"(d_tr), "v"(o_tr), "s"(gbase) : "memory");
        asm volatile("global_load_async_to_lds_b64 %0, %1, %2"
                     :: "v"(d_bl), "v"(o_bl), "s"(gbase) : "memory");
        asm volatile("global_load_async_to_lds_b64 %0, %1, %2"
                     :: "v"(d_br), "v"(o_br), "s"(gbase) : "memory");
    };

    float acc[PW][CPT];
#pragma unroll
    for (int i = 0; i < PW; ++i)
#pragma unroll
        for (int j = 0; j < CPT; ++j)
            acc[i][j] = -__builtin_inff();

    // Pipeline prologue: keep PF groups (PF*4 async loads) in flight.
#pragma unroll
    for (int p = 0; p < PF; ++p)
        issue(p, p % NBUF);

#pragma unroll
    for (int p = 0; p < NP; ++p) {
        if (p + PF < NP) {
            issue(p + PF, (p + PF) % NBUF);
            wait_async(4 * PF);             // group p guaranteed complete
        } else {
            wait_async(4 * (NP - 1 - p));   // tail: 8, 4, 0
        }

        const int gy = 2 * pyr + (p / CW);
        const int gx = p % CW;
        const float in_y = oy + sy * (float)gy;
        const float in_x = ox + sx * (float)gx;
        const float fy = floorf(in_y);
        const float fx = floorf(in_x);
        const float ly = in_y - fy;
        const float lx = in_x - fx;
        const bool valid = (in_y >= 0.0f) & (in_y <= (float)(FH - 1)) &
                           (in_x >= 0.0f) & (in_x <= (float)(FW - 1));
        const float wtl = (1.0f - ly) * (1.0f - lx);
        const float wtr = (1.0f - ly) * lx;
        const float wbl = ly * (1.0f - lx);
        const float wbr = ly * lx;

        const int buf = p % NBUF;
        const int px  = gx >> 1;
#pragma unroll
        for (int j = 0; j < CPT; ++j) {
            const float tl = sh[buf][0][c0 + j];
            const float tr = sh[buf][1][c0 + j];
            const float bl = sh[buf][2][c0 + j];
            const float br = sh[buf][3][c0 + j];
            float v = fmaf(tl, wtl, fmaf(tr, wtr, fmaf(bl, wbl, br * wbr)));
            v = valid ? v : 0.0f;
            acc[px][j] = fmaxf(acc[px][j], v);
        }
    }

    float* dst = out + ((size_t)(roi * PH + pyr)) * (size_t)(PW * FC) + c0;
#pragma unroll
    for (int px = 0; px < PW; ++px) {
        float2 v;
        v.x = acc[px][0];
        v.y = acc[px][1];
        *reinterpret_cast<float2*>(dst + px * FC) = v;
    }
}

extern "C" void kernel_launch(void* const* d_in, const int* in_sizes, int n_in,
                              void* d_out, int out_size, void* d_ws, size_t ws_size,
                              hipStream_t stream) {
    const float* feature = (const float*)d_in[0];
    const float* rois    = (const float*)d_in[1];
    const int*   imgsz   = (const int*)d_in[2];
    float* out = (float*)d_out;

    const int nroi = in_sizes[1] / 4;
    dim3 grid(nroi * PH), block(THREADS);
    hipLaunchKernelGGL(roi_pool_async_kernel, grid, block, 0, stream,
                       feature, rois, imgsz, out);
}